// DecoderGATLayer_40999757807675
// MI455X (gfx1250) — compile-verified
//
#include <hip/hip_runtime.h>

typedef float v2f __attribute__((ext_vector_type(2)));
typedef float v8f __attribute__((ext_vector_type(8)));

#define S_ 192
#define B_ 2
#define E_ 512
#define D_ 256
#define BH_ 4

// ---------------------------------------------------------------------------
// Generic  Y[M,N] = X[M,K] @ W[N,K]^T + bias[N]   (all f32 row-major)
// One wave per 16x16 output tile, V_WMMA_F32_16X16X4_F32 over the K loop.
// A layout (16x4, MxK): lane = (k/2)*16 + m, vgpr = k%2  -> lane m (=lane&15)
//   holds K = (lane>>4)*2 + {0,1} as a contiguous float2.
// B layout (4x16, KxN): symmetric, lane n (=lane&15) holds K=(lane>>4)*2+{0,1}.
// C/D layout (16x16): vgpr r -> M = r + 8*(lane>>4), N = lane&15.
// ---------------------------------------------------------------------------
__global__ void gemm_xwt_kernel(const float* __restrict__ X, const float* __restrict__ W,
                                const float* __restrict__ bias, float* __restrict__ Y,
                                int M, int N, int K) {
  int gw   = (int)((blockIdx.x * blockDim.x + threadIdx.x) >> 5);
  int lane = threadIdx.x & 31;
  int tilesN = N >> 4;
  int total  = (M >> 4) * tilesN;
  if (gw >= total) return;                   // whole-wave uniform exit
  int tm   = (gw / tilesN) << 4;
  int tn   = (gw % tilesN) << 4;
  int half = lane >> 4;
  int l16  = lane & 15;
  const float* arow = X + (size_t)(tm + l16) * K + (half << 1);
  const float* brow = W + (size_t)(tn + l16) * K + (half << 1);
  v8f acc = {};
  for (int k = 0; k < K; k += 4) {
    v2f a = *(const v2f*)(arow + k);
    v2f b = *(const v2f*)(brow + k);
    acc = __builtin_amdgcn_wmma_f32_16x16x4_f32(false, a, false, b,
                                                (short)0, acc, false, false);
  }
  float bv = bias[tn + l16];
#pragma unroll
  for (int r = 0; r < 8; ++r)
    Y[(size_t)(tm + (half << 3) + r) * N + tn + l16] = acc[r] + bv;
}

// ---------------------------------------------------------------------------
// Rank-1 score precomputation (Wa is 1x768 => all score terms are dots):
//   sq[bh*192+t] = dot( (h? e_ : q_)[t,b,:], Wa[0:256]   )
//   sk[bh*192+s] = dot( (h? e_ : k_)[s,b,:], Wa[256:512] )
//   P[t*4+b*2+h] = dot( entity[t,b,h*256:+256], Wa[512:768] )
// 2304 scalar outputs, one thread each (latency-bound, negligible cost).
// ---------------------------------------------------------------------------
__global__ void precompute_kernel(const float* __restrict__ q_, const float* __restrict__ k_,
                                  const float* __restrict__ e_, const float* __restrict__ entity,
                                  const float* __restrict__ Wa,
                                  float* __restrict__ sq, float* __restrict__ sk,
                                  float* __restrict__ P) {
  int idx = blockIdx.x * blockDim.x + threadIdx.x;
  if (idx < 768) {
    int bh = idx / S_, t = idx % S_;
    int b = bh >> 1, h = bh & 1;
    const float* row = (h ? e_ : q_) + ((size_t)t * 2 + b) * D_;
    float acc = 0.f;
    for (int d = 0; d < D_; ++d) acc += row[d] * Wa[d];
    sq[idx] = acc;
  } else if (idx < 1536) {
    int r = idx - 768;
    int bh = r / S_, s = r % S_;
    int b = bh >> 1, h = bh & 1;
    const float* row = (h ? e_ : k_) + ((size_t)s * 2 + b) * D_;
    float acc = 0.f;
    for (int d = 0; d < D_; ++d) acc += row[d] * Wa[D_ + d];
    sk[r] = acc;
  } else if (idx < 2304) {
    int r = idx - 1536;           // r = t*4 + b*2 + half
    int t = r >> 2, b = (r >> 1) & 1, hf = r & 1;
    const float* row = entity + ((size_t)t * 2 + b) * E_ + hf * D_;
    float acc = 0.f;
    for (int d = 0; d < D_; ++d) acc += row[d] * Wa[2 * D_ + d];
    P[r] = acc;
  }
}

__device__ __forceinline__ float wave_max32(float x) {
  for (int o = 16; o > 0; o >>= 1) x = fmaxf(x, __shfl_xor(x, o, 32));
  return x;
}
__device__ __forceinline__ float wave_sum32(float x) {
  for (int o = 16; o > 0; o >>= 1) x += __shfl_xor(x, o, 32);
  return x;
}

// ---------------------------------------------------------------------------
// score[b',t',s'] = sq[s'%4][b'*48+t'/4] + sk[t'%4][s']
//                 + P[i*4 + s'%4] - P[j*4 + s'%4] + ba,   j=(t'%4)*48+s'/4
// then leaky_relu(0.01) and softmax over s' (192).  One wave per (b',t') row,
// 6 elements per lane, wave32 shuffle reductions (no LDS, no barriers).
// ---------------------------------------------------------------------------
__global__ void scores_softmax_kernel(const float* __restrict__ sq, const float* __restrict__ sk,
                                      const float* __restrict__ P, const float* __restrict__ ba,
                                      float* __restrict__ AW) {
  int gw   = (int)((blockIdx.x * blockDim.x + threadIdx.x) >> 5);
  int lane = threadIdx.x & 31;
  if (gw >= BH_ * S_) return;
  int bp = gw / S_, t = gw % S_;
  int i    = bp * 48 + (t >> 2);
  int tmod = t & 3;
  float bav = ba[0];
  float loc[6];
  float m = -3.4e38f;
#pragma unroll
  for (int u = 0; u < 6; ++u) {
    int s    = lane + (u << 5);
    int smod = s & 3;
    int j    = tmod * 48 + (s >> 2);
    float sc = sq[smod * S_ + i] + sk[tmod * S_ + s]
             + P[i * 4 + smod] - P[j * 4 + smod] + bav;
    sc = sc > 0.f ? sc : 0.01f * sc;           // leaky_relu, slope 0.01
    loc[u] = sc;
    m = fmaxf(m, sc);
  }
  m = wave_max32(m);
  float sum = 0.f;
#pragma unroll
  for (int u = 0; u < 6; ++u) { loc[u] = __expf(loc[u] - m); sum += loc[u]; }
  sum = wave_sum32(sum);
  float inv = 1.0f / sum;
  float* row = AW + ((size_t)bp * S_ + t) * S_;
#pragma unroll
  for (int u = 0; u < 6; ++u) row[lane + (u << 5)] = loc[u] * inv;
}

// ---------------------------------------------------------------------------
// out[b'] = AW[b'] (192x192) @ V_head(b') (192x256);  V_head: h==0 ? v : e,
// rows live at (s*2+b)*256.  Stored straight into the merged [s][b][h*256+d]
// layout expected by the output projection.  One wave per 16x16 tile, WMMA.
// ---------------------------------------------------------------------------
__global__ void attn_av_kernel(const float* __restrict__ AW, const float* __restrict__ v_,
                               const float* __restrict__ e_, float* __restrict__ attn) {
  int gw   = (int)((blockIdx.x * blockDim.x + threadIdx.x) >> 5);
  int lane = threadIdx.x & 31;
  if (gw >= BH_ * 12 * 16) return;
  int bp  = gw / (12 * 16);
  int rem = gw % (12 * 16);
  int tm  = (rem / 16) << 4;
  int tn  = (rem % 16) << 4;
  int b = bp >> 1, h = bp & 1;
  const float* V = h ? e_ : v_;
  int half = lane >> 4, l16 = lane & 15;
  const float* arow = AW + ((size_t)bp * S_ + tm + l16) * S_ + (half << 1);
  v8f acc = {};
  for (int k = 0; k < S_; k += 4) {
    v2f a = *(const v2f*)(arow + k);
    int s0 = k + (half << 1);
    v2f bv;
    bv.x = V[(size_t)(s0 * 2 + b) * D_ + tn + l16];
    bv.y = V[(size_t)((s0 + 1) * 2 + b) * D_ + tn + l16];
    acc = __builtin_amdgcn_wmma_f32_16x16x4_f32(false, a, false, bv,
                                                (short)0, acc, false, false);
  }
#pragma unroll
  for (int r = 0; r < 8; ++r) {
    int t = tm + (half << 3) + r;
    attn[(size_t)(t * 2 + b) * E_ + h * D_ + tn + l16] = acc[r];
  }
}

// ---------------------------------------------------------------------------
extern "C" void kernel_launch(void* const* d_in, const int* in_sizes, int n_in,
                              void* d_out, int out_size, void* d_ws, size_t ws_size,
                              hipStream_t stream) {
  const float* query  = (const float*)d_in[0];
  const float* entity = (const float*)d_in[1];
  // d_in[2],[3] (ex_entity/ex_relation) and d_in[12..15] (Wxe/Wxr) feed only
  // the `del we` dead branch of the reference -> intentionally unused.
  const float* Wq = (const float*)d_in[4];  const float* bq = (const float*)d_in[5];
  const float* Wk = (const float*)d_in[6];  const float* bk = (const float*)d_in[7];
  const float* Wv = (const float*)d_in[8];  const float* bv = (const float*)d_in[9];
  const float* We = (const float*)d_in[10]; const float* be = (const float*)d_in[11];
  const float* Wa = (const float*)d_in[16]; const float* ba = (const float*)d_in[17];
  const float* Wo = (const float*)d_in[18]; const float* bo = (const float*)d_in[19];
  float* out = (float*)d_out;

  float* ws   = (float*)d_ws;
  float* q_   = ws;               // 384*256
  float* k_   = q_ + 98304;
  float* v_   = k_ + 98304;
  float* e_   = v_ + 98304;
  float* sq   = e_ + 98304;       // 768
  float* sk   = sq + 768;         // 768
  float* P    = sk + 768;         // 768
  float* AW   = P + 768;          // 4*192*192 = 147456
  float* attn = AW + 147456;      // 384*512   = 196608  (total ~2.96 MB)

  dim3 blk(256);
  // 4 projection GEMMs: [384,512] @ [512,256]^T, 384 tiles = 48 blocks x 8 waves
  hipLaunchKernelGGL(gemm_xwt_kernel, dim3(48), blk, 0, stream, query,  Wq, bq, q_, 384, 256, 512);
  hipLaunchKernelGGL(gemm_xwt_kernel, dim3(48), blk, 0, stream, query,  Wk, bk, k_, 384, 256, 512);
  hipLaunchKernelGGL(gemm_xwt_kernel, dim3(48), blk, 0, stream, query,  Wv, bv, v_, 384, 256, 512);
  hipLaunchKernelGGL(gemm_xwt_kernel, dim3(48), blk, 0, stream, entity, We, be, e_, 384, 256, 512);
  // rank-1 score terms (2304 threads)
  hipLaunchKernelGGL(precompute_kernel, dim3(9), blk, 0, stream, q_, k_, e_, entity, Wa, sq, sk, P);
  // scores + leaky_relu + softmax: 768 rows, one wave each -> 96 blocks
  hipLaunchKernelGGL(scores_softmax_kernel, dim3(96), blk, 0, stream, sq, sk, P, ba, AW);
  // attention * values: 768 tiles -> 96 blocks
  hipLaunchKernelGGL(attn_av_kernel, dim3(96), blk, 0, stream, AW, v_, e_, attn);
  // output projection: [384,512] @ [512,512]^T -> d_out, 768 tiles
  hipLaunchKernelGGL(gemm_xwt_kernel, dim3(96), blk, 0, stream, attn, Wo, bo, out, 384, 512, 512);
}